// TemporalAttention_19670950216591
// MI455X (gfx1250) — compile-verified
//
#include <hip/hip_runtime.h>
#include <hip/hip_bf16.h>

typedef float v2f __attribute__((ext_vector_type(2)));
typedef float v8f __attribute__((ext_vector_type(8)));

#define NHEADS 16
#define HD     64
#define WAVES  8          // waves per block (wave32), one token per wave

__global__ __launch_bounds__(32 * WAVES)
void temporal_head_attn_wmma(const float* __restrict__ q,
                             const float* __restrict__ k,
                             const float* __restrict__ v,
                             float* __restrict__ out,
                             int ntokens)
{
    __shared__ float attn_lds[WAVES][NHEADS * NHEADS];   // 8 KB total

    const int lane = threadIdx.x & 31;
    const int wave = threadIdx.x >> 5;
    const int token = blockIdx.x * WAVES + wave;
    if (token >= ntokens) return;               // wave-uniform guard

    const long base = (long)token * (NHEADS * HD);
    const float* qt = q + base;
    const float* kt = k + base;
    const float* vt = v + base;
    float*       ot = out + base;

    const int mn = lane & 15;    // row m for A-operands, col n for B/C/D
    const int h  = lane >> 4;    // lane-half selects K (A/B) or +8 rows (C/D)

    // ---------------- scores = (qh · khᵀ) / sqrt(64) ----------------
    // A frag (16x4 f32): lane holds q[m][4c+2h+j], j=0,1  -> one float2
    // B frag (4x16 f32): lane holds k[n][4c+2h+j]         -> same pattern
    v8f acc = {};
    #pragma unroll
    for (int c = 0; c < 16; ++c) {
        const int off = 4 * c + 2 * h;
        v2f a = *(const v2f*)(qt + mn * HD + off);
        v2f b = *(const v2f*)(kt + mn * HD + off);
        acc = __builtin_amdgcn_wmma_f32_16x16x4_f32(
                  false, a, false, b, (short)0, acc, false, false);
    }

    // ---------------- causal mask over heads + row softmax ----------------
    // C layout: reg r, lane-half h -> element (m = r + 8h, n = mn).
    // Row m occupies one 16-lane half -> reduce with xor shuffles 1,2,4,8.
    float p[8];
    #pragma unroll
    for (int r = 0; r < 8; ++r) {
        const int m = r + 8 * h;
        float s = acc[r] * 0.125f;               // 1/sqrt(64)
        if (mn > m) s = -1.0e30f;                // tril mask
        float mx = s;
        mx = fmaxf(mx, __shfl_xor(mx, 1));
        mx = fmaxf(mx, __shfl_xor(mx, 2));
        mx = fmaxf(mx, __shfl_xor(mx, 4));
        mx = fmaxf(mx, __shfl_xor(mx, 8));
        float e = __expf(s - mx);
        float sum = e;
        sum += __shfl_xor(sum, 1);
        sum += __shfl_xor(sum, 2);
        sum += __shfl_xor(sum, 4);
        sum += __shfl_xor(sum, 8);
        p[r] = e / sum;
    }

    // ---------------- re-layout attn (C/D layout -> A layout) via LDS ----
    float* al = attn_lds[wave];
    #pragma unroll
    for (int r = 0; r < 8; ++r)
        al[(r + 8 * h) * NHEADS + mn] = p[r];
    __asm__ volatile("s_wait_dscnt 0" ::: "memory");   // single-wave DS RAW

    v2f af[4];
    #pragma unroll
    for (int c = 0; c < 4; ++c)
        af[c] = *(const v2f*)(al + mn * NHEADS + 4 * c + 2 * h);

    // ---------------- out = attn (16x16) · vh (16x64) ----------------
    // N tiled 4x16; B frag: lane holds v[(4c+2h+j)*64 + 16t + n]
    #pragma unroll
    for (int t = 0; t < 4; ++t) {
        v8f d = {};
        #pragma unroll
        for (int c = 0; c < 4; ++c) {
            v2f b;
            b.x = vt[(4 * c + 2 * h + 0) * HD + 16 * t + mn];
            b.y = vt[(4 * c + 2 * h + 1) * HD + 16 * t + mn];
            d = __builtin_amdgcn_wmma_f32_16x16x4_f32(
                    false, af[c], false, b, (short)0, d, false, false);
        }
        #pragma unroll
        for (int r = 0; r < 8; ++r)
            ot[(r + 8 * h) * HD + 16 * t + mn] = d[r];
    }
}

extern "C" void kernel_launch(void* const* d_in, const int* in_sizes, int n_in,
                              void* d_out, int out_size, void* d_ws, size_t ws_size,
                              hipStream_t stream)
{
    const float* q = (const float*)d_in[0];
    const float* k = (const float*)d_in[1];
    const float* v = (const float*)d_in[2];
    float* out = (float*)d_out;

    const int ntokens = in_sizes[0] / (NHEADS * HD);        // B*S = 32768
    const int blocks  = (ntokens + WAVES - 1) / WAVES;      // 4096

    temporal_head_attn_wmma<<<blocks, 32 * WAVES, 0, stream>>>(q, k, v, out, ntokens);
}